// LineEvoLayer_773094113319
// MI455X (gfx1250) — compile-verified
//
#include <hip/hip_runtime.h>
#include <math.h>

#define N_NODES  50000
#define N_EDGES  800000
#define DIM      128
#define N_MOLS   2048
#define RBF_BINS 6

typedef float v2f __attribute__((ext_vector_type(2)));
typedef float v8f __attribute__((ext_vector_type(8)));

// ---------------------------------------------------------------------------
// Kernel 1: h = x @ W.T + b  via V_WMMA_F32_16X16X4_F32 (full fp32 WMMA).
// grid.x = 3125 row tiles (16 rows each), block = 256 threads = 8 waves,
// wave w computes the 16x16 tile at columns [16w, 16w+16).
// A-layout (16x4 f32): lane l<16 -> A[M=l][K=0..1] in v0,v1; lane>=16 -> K=2..3.
// B-layout (4x16 f32): lane l<16 -> B[K=0..1][N=l]; lane>=16 -> K=2..3.
//   B[k][n] = W[n][k], so each lane loads a contiguous float2 from row (col0+l) of W.
// ---------------------------------------------------------------------------
__global__ __launch_bounds__(256) void gemm_h_kernel(
    const float* __restrict__ x, const float* __restrict__ W,
    const float* __restrict__ b, float* __restrict__ h) {
  const int lane = threadIdx.x & 31;
  const int wave = threadIdx.x >> 5;
  const int row0 = blockIdx.x * 16;
  const int col0 = wave * 16;
  const int half = lane >> 4;   // 0: K pair {0,1}, 1: K pair {2,3}
  const int l15  = lane & 15;

  const float* xrow = x + (size_t)(row0 + l15) * DIM + half * 2;
  const float* wrow = W + (size_t)(col0 + l15) * DIM + half * 2;

  v8f acc = {};
#pragma unroll
  for (int k = 0; k < DIM; k += 4) {
    v2f a  = *(const v2f*)(xrow + k);
    v2f bm = *(const v2f*)(wrow + k);
    acc = __builtin_amdgcn_wmma_f32_16x16x4_f32(
        /*neg_a=*/false, a, /*neg_b=*/false, bm,
        /*c_mod=*/(short)0, acc, /*reuse_a=*/false, /*reuse_b=*/false);
  }

  const float bias = b[col0 + l15];
  // D layout: VGPR r -> lanes 0-15: (M=r, N=lane); lanes 16-31: (M=8+r, N=lane-16)
#pragma unroll
  for (int r = 0; r < 8; ++r) {
    h[(size_t)(row0 + r + half * 8) * DIM + col0 + l15] = acc[r] + bias;
  }
}

// ---------------------------------------------------------------------------
// Kernel 2: init molecule accumulator region (sum -> 0, max -> -inf)
// ---------------------------------------------------------------------------
__global__ void init_mol_kernel(float* __restrict__ mol) {
  int idx = blockIdx.x * 256 + threadIdx.x;          // 2048 * 256
  int col = idx & 255;
  mol[idx] = (col < DIM) ? 0.0f : -INFINITY;
}

// ---------------------------------------------------------------------------
// float atomic-max via signed-max / unsigned-min trick
// ---------------------------------------------------------------------------
__device__ __forceinline__ void atomicMaxFloat(float* addr, float val) {
  if (val >= 0.0f) {
    atomicMax((int*)addr, __float_as_int(val));
  } else {
    atomicMin((unsigned int*)addr, __float_as_uint(val));
  }
}

__device__ __forceinline__ float elu1(float v) {
  return v > 0.0f ? v : (__expf(v) - 1.0f);
}

// ---------------------------------------------------------------------------
// Kernel 3: edge phase. One wave32 per edge, 4 dims per lane (float4 = b128
// gathers out of the L2-resident h table). Gate dot reduced with wave32
// shfl_xor; scatter with global float atomics into the molecule block.
// ---------------------------------------------------------------------------
__global__ __launch_bounds__(256) void edge_kernel(
    const float* __restrict__ h, const float* __restrict__ pos,
    const int* __restrict__ edges, const int* __restrict__ batch,
    const float* __restrict__ attn, const float* __restrict__ W_rbf,
    const float* __restrict__ Wg, const float* __restrict__ bg,
    float* __restrict__ atom_out, float* __restrict__ pos_out,
    int* __restrict__ batch_out, float* __restrict__ mol) {
  const int lane = threadIdx.x & 31;
  const int wave = threadIdx.x >> 5;
  const int e = blockIdx.x * 8 + wave;
  if (e >= N_EDGES) return;

  const int src = edges[2 * e + 0];
  const int dst = edges[2 * e + 1];

  // distance (redundant per lane; 3 floats each, cheap)
  const float dx = pos[3 * dst + 0] - pos[3 * src + 0];
  const float dy = pos[3 * dst + 1] - pos[3 * src + 1];
  const float dz = pos[3 * dst + 2] - pos[3 * src + 2];
  float dist = sqrtf(dx * dx + dy * dy + dz * dz);
  dist = fmaxf(dist, 0.1f);

  // RBF: centers = linspace(0,5,6) = 0,1,2,3,4,5
  float rb[RBF_BINS];
#pragma unroll
  for (int j = 0; j < RBF_BINS; ++j) {
    const float t = dist - (float)j;
    rb[j] = __expf(-t * t);
  }

  const int d0 = lane * 4;
  const float4 hs = *(const float4*)(h + (size_t)src * DIM + d0);
  const float4 hd = *(const float4*)(h + (size_t)dst * DIM + d0);
  const float hsv[4] = {hs.x, hs.y, hs.z, hs.w};
  const float hdv[4] = {hd.x, hd.y, hd.z, hd.w};

  float at[4];
  float p = 0.0f;  // partial gate dot
#pragma unroll
  for (int c = 0; c < 4; ++c) {
    const int d = d0 + c;
    float ev = elu1(hsv[c] + hdv[c]);
    float m = 0.0f;
#pragma unroll
    for (int j = 0; j < RBF_BINS; ++j) m += rb[j] * W_rbf[d * RBF_BINS + j];
    const float a = elu1(ev * m * attn[d]);
    at[c] = a;
    p += a * Wg[d];
  }

  // wave32 reduction for the gate score
#pragma unroll
  for (int off = 16; off > 0; off >>= 1) p += __shfl_xor(p, off, 32);
  const float score = 1.0f / (1.0f + __expf(-(p + bg[0])));

  // outputs
  float4 av = make_float4(at[0], at[1], at[2], at[3]);
  *(float4*)(atom_out + (size_t)e * DIM + d0) = av;
  if (lane < 3) {
    pos_out[(size_t)e * 3 + lane] = 0.5f * (pos[3 * src + lane] + pos[3 * dst + lane]);
  }
  const int molid = batch[src];
  if (lane == 0) batch_out[e] = molid;

  // segment sum (gated) + segment max
  float* mrow = mol + (size_t)molid * 256;
#pragma unroll
  for (int c = 0; c < 4; ++c) {
    const int d = d0 + c;
    atomicAdd(mrow + d, at[c] * score);
    atomicMaxFloat(mrow + DIM + d, at[c]);
  }
}

// ---------------------------------------------------------------------------
// Host launcher
// ---------------------------------------------------------------------------
extern "C" void kernel_launch(void* const* d_in, const int* in_sizes, int n_in,
                              void* d_out, int out_size, void* d_ws, size_t ws_size,
                              hipStream_t stream) {
  (void)in_sizes; (void)n_in; (void)out_size; (void)ws_size;

  const float* x     = (const float*)d_in[0];
  const float* pos   = (const float*)d_in[1];
  const int*   edges = (const int*)d_in[2];
  const int*   batch = (const int*)d_in[3];
  const float* W     = (const float*)d_in[4];
  const float* b     = (const float*)d_in[5];
  const float* attn  = (const float*)d_in[6];
  const float* W_rbf = (const float*)d_in[7];
  const float* Wg    = (const float*)d_in[8];
  const float* bg    = (const float*)d_in[9];

  // d_out layout (flat, return order): atom_repr | pos_out | batch_e | mol_repr
  float* base      = (float*)d_out;
  float* atom_out  = base;                                   // 800000*128 f32
  float* pos_out   = base + (size_t)N_EDGES * DIM;           // 800000*3  f32
  int*   batch_out = (int*)(pos_out + (size_t)N_EDGES * 3);  // 800000    i32
  float* mol       = (float*)(batch_out + N_EDGES);          // 2048*256  f32

  float* h = (float*)d_ws;  // 50000*128 f32 = 25.6 MB scratch

  // Phase 1: fp32 WMMA GEMM h = x @ W.T + b
  gemm_h_kernel<<<N_NODES / 16, 256, 0, stream>>>(x, W, b, h);

  // Phase 2: init molecule accumulators (sum=0, max=-inf)
  init_mol_kernel<<<N_MOLS, 256, 0, stream>>>(mol);

  // Phase 3: edge streaming + scatter
  edge_kernel<<<N_EDGES / 8, 256, 0, stream>>>(
      h, pos, edges, batch, attn, W_rbf, Wg, bg,
      atom_out, pos_out, batch_out, mol);
}